// GAT_14233521619312
// MI455X (gfx1250) — compile-verified
//
#include <hip/hip_runtime.h>
#include <hip/hip_bf16.h>
#include <math.h>

typedef __attribute__((ext_vector_type(2))) float v2f;
typedef __attribute__((ext_vector_type(8))) float v8f;

#define N_NODES 100000
#define N_EDGES 1600000
#define IN_DIM  128
#define HD      128     // H*D (layer-1 output width)
#define H1      4
#define D1      32
#define OUT_DIM 32
#define SLOPE   0.2f

// -------- helpers --------------------------------------------------------

__device__ __forceinline__ void atomicMaxF(float* addr, float v) {
    // sign-split trick: monotone in int-bits for >=0, reversed in uint-bits for <0
    if (v >= 0.0f) atomicMax((int*)addr, __float_as_int(v));
    else           atomicMin((unsigned int*)addr, __float_as_uint(v));
}

// -------- WMMA GEMM: C[M,Nc] = A[M,K] @ B[K,Nc], one wave per 16x16 tile --

__global__ void gemm16_wmma(const float* __restrict__ A,
                            const float* __restrict__ B,
                            float* __restrict__ C,
                            int M, int Ncols, int K) {
    const int wave = blockIdx.x * (blockDim.x >> 5) + (threadIdx.x >> 5);
    const int tiles_n = Ncols >> 4;
    const int tm = wave / tiles_n;
    const int tn = wave % tiles_n;
    if (tm * 16 >= M) return;                   // wave-uniform guard (EXEC stays all-1)

    const int lane = threadIdx.x & 31;
    const int lo = lane & 15;                   // M index (A) / N index (B,C)
    const int hi = lane >> 4;                   // K-half select

    const float* arow = A + (size_t)(tm * 16 + lo) * K;
    const float* bcol = B + tn * 16 + lo;

    v8f c = {0.f, 0.f, 0.f, 0.f, 0.f, 0.f, 0.f, 0.f};
    for (int k = 0; k < K; k += 4) {
        v2f a, b;
        // A 16x4: lanes 0-15 -> K = k+0,k+1 ; lanes 16-31 -> K = k+2,k+3
        a.x = arow[k + 2 * hi];
        a.y = arow[k + 2 * hi + 1];
        // B 4x16: same K split, N striped across lanes
        b.x = bcol[(size_t)(k + 2 * hi) * Ncols];
        b.y = bcol[(size_t)(k + 2 * hi + 1) * Ncols];
        c = __builtin_amdgcn_wmma_f32_16x16x4_f32(false, a, false, b,
                                                  (short)0, c, false, false);
    }
    // C/D layout: VGPR v, lanes 0-15 -> M = v ; lanes 16-31 -> M = v+8
    float* crow = C + (size_t)(tm * 16) * Ncols + tn * 16 + lo;
#pragma unroll
    for (int v = 0; v < 8; ++v)
        crow[(size_t)(v + 8 * hi) * Ncols] = c[v];
}

// -------- attention score dots: el/er[n,h] = <feat[n,h,:], al/ar[h,:]> ----

__global__ void attn_scores(const float* __restrict__ feat,
                            const float* __restrict__ al,
                            const float* __restrict__ ar,
                            float* __restrict__ el, float* __restrict__ er,
                            int n_nodes, int heads, int d) {
    int t = blockIdx.x * blockDim.x + threadIdx.x;
    if (t >= n_nodes * heads) return;
    int n = t / heads, h = t % heads;
    const float* f  = feat + (size_t)n * heads * d + h * d;
    const float* a1 = al + h * d;
    const float* a2 = ar + h * d;
    float s1 = 0.f, s2 = 0.f;
    for (int i = 0; i < d; ++i) { float x = f[i]; s1 += x * a1[i]; s2 += x * a2[i]; }
    el[t] = s1; er[t] = s2;
}

__global__ void fill_f32(float* __restrict__ p, float v, int n) {
    int t = blockIdx.x * blockDim.x + threadIdx.x;
    if (t < n) p[t] = v;
}

// -------- edge pass 1: e = leakyrelu(el[src]+er[dst]); segment max by dst -

__global__ void edge_max(const int* __restrict__ src, const int* __restrict__ dst,
                         const float* __restrict__ el, const float* __restrict__ er,
                         float* __restrict__ ebuf, float* __restrict__ m,
                         int n_edges, int heads) {
    int t = blockIdx.x * blockDim.x + threadIdx.x;
    if (t >= n_edges * heads) return;
    int e = t / heads, h = t % heads;
    float v = el[src[e] * heads + h] + er[dst[e] * heads + h];
    v = v > 0.f ? v : SLOPE * v;
    ebuf[t] = v;
    atomicMaxF(&m[dst[e] * heads + h], v);
}

// -------- edge pass 2: ex = exp(e - m[dst]); segment sum by dst -----------

__global__ void edge_expsum(const int* __restrict__ dst,
                            float* __restrict__ ebuf,
                            const float* __restrict__ m,
                            float* __restrict__ denom,
                            int n_edges, int heads) {
    int t = blockIdx.x * blockDim.x + threadIdx.x;
    if (t >= n_edges * heads) return;
    int e = t / heads, h = t % heads;
    int di = dst[e] * heads + h;
    float ex = __expf(ebuf[t] - m[di]);
    ebuf[t] = ex;
    atomicAdd(&denom[di], ex);
}

// -------- edge pass 3 (layer 1): wave per edge, float4 per lane -----------

__global__ void edge_agg_l1(const int* __restrict__ src, const int* __restrict__ dst,
                            const float* __restrict__ feat,
                            const float* __restrict__ ebuf,
                            const float* __restrict__ denom,
                            float* __restrict__ agg, int n_edges) {
    int wave = blockIdx.x * (blockDim.x >> 5) + (threadIdx.x >> 5);
    if (wave >= n_edges) return;
    int lane = threadIdx.x & 31;
    int s = src[wave], d = dst[wave];
    int head = lane >> 3;                         // (lane*4)/32
    float alpha = ebuf[wave * H1 + head] / denom[d * H1 + head];
    const float4 f = *(const float4*)(feat + (size_t)s * HD + lane * 4);
    float* ap = agg + (size_t)d * HD + lane * 4;
    atomicAdd(ap + 0, alpha * f.x);
    atomicAdd(ap + 1, alpha * f.y);
    atomicAdd(ap + 2, alpha * f.z);
    atomicAdd(ap + 3, alpha * f.w);
}

// -------- edge pass 3 (layer 2): wave per edge, 1 float per lane ----------

__global__ void edge_agg_l2(const int* __restrict__ src, const int* __restrict__ dst,
                            const float* __restrict__ feat,
                            const float* __restrict__ ebuf,
                            const float* __restrict__ denom,
                            float* __restrict__ agg, int n_edges) {
    int wave = blockIdx.x * (blockDim.x >> 5) + (threadIdx.x >> 5);
    if (wave >= n_edges) return;
    int lane = threadIdx.x & 31;
    int s = src[wave], d = dst[wave];
    float alpha = ebuf[wave] / denom[d];
    float f = feat[(size_t)s * OUT_DIM + lane];
    atomicAdd(agg + (size_t)d * OUT_DIM + lane, alpha * f);
}

// -------- epilogues -------------------------------------------------------

__global__ void bias_elu(float* __restrict__ x, const float* __restrict__ b,
                         int total, int f) {
    int t = blockIdx.x * blockDim.x + threadIdx.x;
    if (t >= total) return;
    float v = x[t] + b[t % f];
    x[t] = v > 0.f ? v : expm1f(v);
}

__global__ void bias_out(const float* __restrict__ agg, const float* __restrict__ b,
                         float* __restrict__ out, int total, int f) {
    int t = blockIdx.x * blockDim.x + threadIdx.x;
    if (t >= total) return;
    out[t] = agg[t] + b[t % f];
}

// -------- launch ----------------------------------------------------------

static inline int cdiv(long long a, long long b) { return (int)((a + b - 1) / b); }

extern "C" void kernel_launch(void* const* d_in, const int* in_sizes, int n_in,
                              void* d_out, int out_size, void* d_ws, size_t ws_size,
                              hipStream_t stream) {
    const float* h   = (const float*)d_in[0];
    const float* W1  = (const float*)d_in[1];
    const float* al1 = (const float*)d_in[2];
    const float* ar1 = (const float*)d_in[3];
    const float* b1  = (const float*)d_in[4];
    const float* W2  = (const float*)d_in[5];
    const float* al2 = (const float*)d_in[6];
    const float* ar2 = (const float*)d_in[7];
    const float* b2  = (const float*)d_in[8];
    const int*   src = (const int*)d_in[9];
    const int*   dst = (const int*)d_in[10];
    float* out = (float*)d_out;

    // workspace layout (float offsets)
    float* w = (float*)d_ws;
    float* feat1 = w;                         // N*128  = 12,800,000
    float* agg1  = feat1 + 12800000;          // N*128  (reused in place as h1)
    float* ebuf  = agg1  + 12800000;          // E*4    =  6,400,000
    float* el1   = ebuf  + 6400000;           // N*4
    float* er1   = el1   + 400000;
    float* m1    = er1   + 400000;
    float* dn1   = m1    + 400000;
    float* feat2 = dn1   + 400000;            // N*32 = 3,200,000
    float* agg2  = feat2 + 3200000;           // N*32
    float* el2   = agg2  + 3200000;           // N
    float* er2   = el2   + 100000;
    float* m2    = er2   + 100000;
    float* dn2   = m2    + 100000;

    const float NEG_INF = -INFINITY;          // 0xff800000 bit pattern (host-side)
    const int BT = 256;                       // 8 waves per block (wave32)

    // ===== layer 1 =====
    // feat1 = h @ W1   : 6250 M-tiles x 8 N-tiles = 50000 waves
    gemm16_wmma<<<cdiv(6250LL * 8, 8), BT, 0, stream>>>(h, W1, feat1,
                                                        N_NODES, HD, IN_DIM);
    attn_scores<<<cdiv((long long)N_NODES * H1, BT), BT, 0, stream>>>(
        feat1, al1, ar1, el1, er1, N_NODES, H1, D1);

    fill_f32<<<cdiv((long long)N_NODES * H1, BT), BT, 0, stream>>>(m1, NEG_INF, N_NODES * H1);
    fill_f32<<<cdiv((long long)N_NODES * H1, BT), BT, 0, stream>>>(dn1, 0.f, N_NODES * H1);
    fill_f32<<<cdiv((long long)N_NODES * HD, BT), BT, 0, stream>>>(agg1, 0.f, N_NODES * HD);

    edge_max<<<cdiv((long long)N_EDGES * H1, BT), BT, 0, stream>>>(
        src, dst, el1, er1, ebuf, m1, N_EDGES, H1);
    edge_expsum<<<cdiv((long long)N_EDGES * H1, BT), BT, 0, stream>>>(
        dst, ebuf, m1, dn1, N_EDGES, H1);
    edge_agg_l1<<<cdiv(N_EDGES, 8), BT, 0, stream>>>(
        src, dst, feat1, ebuf, dn1, agg1, N_EDGES);

    // h1 = elu(agg1 + b1), in place
    bias_elu<<<cdiv((long long)N_NODES * HD, BT), BT, 0, stream>>>(agg1, b1, N_NODES * HD, HD);
    float* h1 = agg1;

    // ===== layer 2 =====
    // feat2 = h1 @ W2  : 6250 M-tiles x 2 N-tiles = 12500 waves
    gemm16_wmma<<<cdiv(6250LL * 2, 8), BT, 0, stream>>>(h1, W2, feat2,
                                                        N_NODES, OUT_DIM, HD);
    attn_scores<<<cdiv((long long)N_NODES, BT), BT, 0, stream>>>(
        feat2, al2, ar2, el2, er2, N_NODES, 1, OUT_DIM);

    fill_f32<<<cdiv((long long)N_NODES, BT), BT, 0, stream>>>(m2, NEG_INF, N_NODES);
    fill_f32<<<cdiv((long long)N_NODES, BT), BT, 0, stream>>>(dn2, 0.f, N_NODES);
    fill_f32<<<cdiv((long long)N_NODES * OUT_DIM, BT), BT, 0, stream>>>(agg2, 0.f, N_NODES * OUT_DIM);

    edge_max<<<cdiv((long long)N_EDGES, BT), BT, 0, stream>>>(
        src, dst, el2, er2, ebuf, m2, N_EDGES, 1);
    edge_expsum<<<cdiv((long long)N_EDGES, BT), BT, 0, stream>>>(
        dst, ebuf, m2, dn2, N_EDGES, 1);
    edge_agg_l2<<<cdiv(N_EDGES, 8), BT, 0, stream>>>(
        src, dst, feat2, ebuf, dn2, agg2, N_EDGES);

    bias_out<<<cdiv((long long)N_NODES * OUT_DIM, BT), BT, 0, stream>>>(
        agg2, b2, out, N_NODES * OUT_DIM, OUT_DIM);
}